// HybridQuanvolutionNet_65481071397926
// MI455X (gfx1250) — compile-verified
//
#include <hip/hip_runtime.h>
#include <math.h>

// ---- CDNA5 WMMA types ----
typedef __attribute__((ext_vector_type(2))) float v2f;
typedef __attribute__((ext_vector_type(8))) float v8f;

#define SPW   16     // samples per wave (M tile)
#define NFEAT 784    // 4*14*14 conv features
#define KTOT  785    // feats + expectation
#define KPAD  788    // padded to multiple of 4; 788%64=20 -> conflict-free A reads

// Build WMMA-B-ready, zero-padded weight tile in workspace:
// fc_wp[n][k] = fc_w[n][k] for n<10, k<785, else 0.   (16 x KPAD floats)
__global__ __launch_bounds__(256)
void prep_fcw_kernel(const float* __restrict__ fc_w, float* __restrict__ fc_wp) {
  const int total = 16 * KPAD;
  for (int t = blockIdx.x * blockDim.x + threadIdx.x; t < total;
       t += gridDim.x * blockDim.x) {
    const int n = t / KPAD;
    const int k = t - n * KPAD;
    fc_wp[t] = (n < 10 && k < KTOT) ? fc_w[n * KTOT + k] : 0.0f;
  }
}

// Saturation-safe fast tanh: 1 - 2/(exp(2x)+1)  (arg is already 2x)
// exp->inf => 1.0 ; exp->0 => -1.0 ; uses v_exp_f32 + v_rcp_f32.
__device__ __forceinline__ float fast_tanh_from_2x(float two_x) {
  const float e = __expf(two_x);
  return fmaf(-2.0f, __builtin_amdgcn_rcpf(e + 1.0f), 1.0f);
}

// One 16x16x4 fp32 WMMA step, branch-free:
// A (16x4): lane l -> row l&15, VGPR0/1 = K = kb+2*(l>>4) + {0,1}  (from LDS)
// B (4x16): lane l -> col l&15, same K                             (from fc_wp)
__device__ __forceinline__ v8f wmma_step(const float* arow, const float* brow,
                                         int ka, v8f acc) {
  const v2f a = *(const v2f*)(arow + ka);
  const v2f b = *(const v2f*)(brow + ka);
  return __builtin_amdgcn_wmma_f32_16x16x4_f32(
      /*neg_a=*/false, a, /*neg_b=*/false, b,
      /*c_mod=*/(short)0, acc, /*reuse_a=*/false, /*reuse_b=*/false);
}

__global__ __launch_bounds__(32)
void quanv_head_kernel(const float* __restrict__ x,       // [B][1][28][28]
                       const float* __restrict__ conv_w,  // [4][1][2][2]
                       const float* __restrict__ conv_b,  // [4]
                       const float* __restrict__ qfc_w,   // scalar
                       const float* __restrict__ qfc_b,   // scalar
                       const float* __restrict__ fc_wp,   // [16][KPAD] padded
                       const float* __restrict__ fc_b,    // [10]
                       float* __restrict__ out,           // [B][10]
                       int B) {
  alignas(16) __shared__ float feats[SPW][KPAD];   // 50432 B
  alignas(16) __shared__ float cmat[16 * 17];      // 1088 B, stride 17

  const int lane = threadIdx.x;        // wave32
  const int lo = lane & 15;
  const int hi = lane >> 4;
  const int base = blockIdx.x * SPW;

  // Uniform small params (L2/WGP$ resident).
  float w00[4], w01[4], w10[4], w11[4], cb[4];
#pragma unroll
  for (int c = 0; c < 4; ++c) {
    w00[c] = conv_w[c * 4 + 0];
    w01[c] = conv_w[c * 4 + 1];
    w10[c] = conv_w[c * 4 + 2];
    w11[c] = conv_w[c * 4 + 3];
    cb[c]  = conv_b[c];
  }
  const float qw2 = 2.0f * qfc_w[0];   // pre-scaled for tanh(2x) form
  const float qb2 = 2.0f * qfc_b[0];

  // ---- Stage 1: fused conv2x2/s2 + tanh-expectation, feats -> LDS ----
  for (int s = 0; s < SPW; ++s) {
    const int sg = base + s;
    if (sg >= B) {
      for (int k = lane; k < KPAD; k += 32) feats[s][k] = 0.0f;
      continue;
    }
    const float* xs = x + (size_t)sg * NFEAT;

    // prefetch next sample's 3136B while we crunch this one
    if (s + 1 < SPW && sg + 1 < B)
      __builtin_prefetch(xs + NFEAT + lane * (NFEAT / 32), 0, 0);

    float tsum = 0.0f;
    for (int p = lane; p < 196; p += 32) {      // 196 = 14*14 positions
      const int i = p / 14, j = p - i * 14;
      const float* r0 = xs + (2 * i) * 28 + 2 * j;   // 8B-aligned
      const v2f top = *(const v2f*)r0;
      const v2f bot = *(const v2f*)(r0 + 28);
#pragma unroll
      for (int c = 0; c < 4; ++c) {
        float f = fmaf(top.x, w00[c],
                  fmaf(top.y, w01[c],
                  fmaf(bot.x, w10[c],
                  fmaf(bot.y, w11[c], cb[c]))));
        feats[s][c * 196 + p] = f;
        tsum += fast_tanh_from_2x(fmaf(f, qw2, qb2));
      }
    }
    // wave reduction of tanh sum -> expectation at K=784
#pragma unroll
    for (int off = 16; off > 0; off >>= 1)
      tsum += __shfl_xor(tsum, off, 32);
    if (lane == 0) feats[s][NFEAT] = tsum * (1.0f / (float)NFEAT);
    if (lane >= 1 && lane < KPAD - NFEAT) feats[s][NFEAT + lane] = 0.0f; // 785..787
  }
  __syncthreads();

  // ---- Stage 2: head GEMM via v_wmma_f32_16x16x4_f32, branch-free loop ----
  const float* arow = &feats[lo][0];           // LDS row for this lane's M
  const float* brow = fc_wp + lo * KPAD;       // padded weight row (N = lo)
  const int koff = 2 * hi;

  v8f acc0 = {};
  v8f acc1 = {};
#pragma unroll 2
  for (int kb = 0; kb < 784; kb += 8) {
    acc0 = wmma_step(arow, brow, kb + koff,     acc0);
    acc1 = wmma_step(arow, brow, kb + 4 + koff, acc1);
  }
  acc0 = wmma_step(arow, brow, 784 + koff, acc0);  // K = 784..787 (zero-padded)
#pragma unroll
  for (int v = 0; v < 8; ++v) acc0[v] += acc1[v];

  // ---- Stage 3: log_softmax epilogue ----
  // C layout: lane l holds N = l&15; VGPR v holds M = v + 8*(l>>4)
#pragma unroll
  for (int v = 0; v < 8; ++v) {
    const int m = v + 8 * hi;
    cmat[m * 17 + lo] = acc0[v];
  }
  __syncthreads();

  if (lane < 16) {
    const int sg = base + lane;
    if (sg < B) {
      float lg[10];
      float mx = -__builtin_inff();
#pragma unroll
      for (int n = 0; n < 10; ++n) {
        lg[n] = cmat[lane * 17 + n] + fc_b[n];
        mx = fmaxf(mx, lg[n]);
      }
      float se = 0.0f;
#pragma unroll
      for (int n = 0; n < 10; ++n) se += __expf(lg[n] - mx);
      const float lse = mx + __logf(se);
      float* o = out + (size_t)sg * 10;
#pragma unroll
      for (int n = 0; n < 10; ++n) o[n] = lg[n] - lse;
    }
  }
}

extern "C" void kernel_launch(void* const* d_in, const int* in_sizes, int n_in,
                              void* d_out, int out_size, void* d_ws, size_t ws_size,
                              hipStream_t stream) {
  const float* x      = (const float*)d_in[0];
  const float* conv_w = (const float*)d_in[1];
  const float* conv_b = (const float*)d_in[2];
  const float* qfc_w  = (const float*)d_in[3];
  const float* qfc_b  = (const float*)d_in[4];
  const float* fc_w   = (const float*)d_in[5];
  const float* fc_b   = (const float*)d_in[6];
  float* out   = (float*)d_out;
  float* fc_wp = (float*)d_ws;               // 16*KPAD*4 = 50432 B scratch

  const int B = in_sizes[0] / NFEAT;         // 65536
  const int grid = (B + SPW - 1) / SPW;      // 4096 single-wave workgroups

  prep_fcw_kernel<<<13, 256, 0, stream>>>(fc_w, fc_wp);
  quanv_head_kernel<<<grid, 32, 0, stream>>>(x, conv_w, conv_b, qfc_w, qfc_b,
                                             fc_wp, fc_b, out, B);
}